// GCNJK_4501125726319
// MI455X (gfx1250) — compile-verified
//
#include <hip/hip_runtime.h>

typedef __attribute__((ext_vector_type(2))) float v2f;
typedef __attribute__((ext_vector_type(8))) float v8f;

#define HDIM 128
#define ODIM 40

// ---------------------------------------------------------------------------
// Degree / normalization
// ---------------------------------------------------------------------------
__global__ void k_init_deg(float* __restrict__ deg, int n) {
    int i = blockIdx.x * blockDim.x + threadIdx.x;
    if (i < n) deg[i] = 1.0f;                 // self-loop contributes 1
}

__global__ void k_edge_deg(const int* __restrict__ col, float* __restrict__ deg, int e) {
    int i = blockIdx.x * blockDim.x + threadIdx.x;
    if (i < e) atomicAdd(&deg[col[i]], 1.0f);
}

__global__ void k_finish_deg(float* __restrict__ deg, int n) {
    int i = blockIdx.x * blockDim.x + threadIdx.x;
    if (i < n) {
        float d = deg[i];
        deg[i] = (d > 0.0f) ? rsqrtf(d) : 0.0f;   // deg >= 1 always, but keep parity
    }
}

__global__ void k_zero4(float4* __restrict__ p, long n4) {
    long i = (long)blockIdx.x * blockDim.x + threadIdx.x;
    if (i < n4) p[i] = make_float4(0.f, 0.f, 0.f, 0.f);
}

// ---------------------------------------------------------------------------
// WMMA fp32 GEMM:  C[M x ncols] = A[M x 128] * B[128 x ncols]  (+ bias)
// One wave per 16-row tile; all N-tiles per wave; B transposed in LDS.
// ---------------------------------------------------------------------------
template <int NPAD>
__global__ __launch_bounds__(256) void k_gemm_wmma(
        const float* __restrict__ A, const float* __restrict__ B,
        const float* __restrict__ bias, float* __restrict__ C,
        int M, int ncols) {
    constexpr int K  = 128;
    constexpr int NT = NPAD / 16;
    constexpr int LSTR = K + 2;                 // even stride: 8B-aligned b64, bank-spread
    __shared__ float ldsBT[NPAD * LSTR];

    // cooperative transpose-load of B into LDS (zero-pad cols >= ncols)
    for (int i = threadIdx.x; i < NPAD * K; i += 256) {
        int k = i & (K - 1);
        int n = i >> 7;
        ldsBT[n * LSTR + k] = (n < ncols) ? B[k * ncols + n] : 0.0f;
    }
    __syncthreads();

    const int lane  = threadIdx.x & 31;
    const int tile  = blockIdx.x * 8 + (threadIdx.x >> 5);
    const int mtile = M >> 4;
    if (tile >= mtile) return;

    const int rowA = tile * 16 + (lane & 15);
    const int koff = (lane >> 4) * 2;           // lanes 0-15: K{0,1}; 16-31: K{2,3}
    const float* aptr = A + (size_t)rowA * K + koff;

    v8f acc[NT];
    v8f z8 = {0.f, 0.f, 0.f, 0.f, 0.f, 0.f, 0.f, 0.f};
#pragma unroll
    for (int nt = 0; nt < NT; ++nt) acc[nt] = z8;

#pragma unroll 4
    for (int k = 0; k < K; k += 4) {
        v2f a = *(const v2f*)(aptr + k);
#pragma unroll
        for (int nt = 0; nt < NT; ++nt) {
            int n = nt * 16 + (lane & 15);
            v2f b = *(const v2f*)(&ldsBT[n * LSTR + k + koff]);
            acc[nt] = __builtin_amdgcn_wmma_f32_16x16x4_f32(
                /*neg_a=*/false, a, /*neg_b=*/false, b,
                /*c_mod=*/(short)0, acc[nt],
                /*reuse_a=*/false, /*reuse_b=*/false);
        }
    }

    // D layout: VGPR r -> row tile*16 + r (+8 for upper half-wave); col = n_base + lane&15
    const int rbase = tile * 16 + ((lane >> 4) << 3);
#pragma unroll
    for (int nt = 0; nt < NT; ++nt) {
        int n = nt * 16 + (lane & 15);
        if (n >= ncols) continue;
        float badd = bias ? bias[n] : 0.0f;
#pragma unroll
        for (int r = 0; r < 8; ++r) {
            C[(size_t)(rbase + r) * ncols + n] = acc[nt][r] + badd;
        }
    }
}

// ---------------------------------------------------------------------------
// Edge scatter: one wave per edge, 4 channels per lane, L2-resident atomics
// ---------------------------------------------------------------------------
__global__ __launch_bounds__(256) void k_scatter(
        const float* __restrict__ h, const float* __restrict__ dis,
        const int* __restrict__ row, const int* __restrict__ col,
        float* __restrict__ y, int e) {
    int w = blockIdx.x * 8 + (threadIdx.x >> 5);
    if (w >= e) return;
    int lane = threadIdx.x & 31;
    int r = row[w];
    int c = col[w];
    float nrm = dis[r] * dis[c];
    const float4 v = *(const float4*)(h + (size_t)r * HDIM + lane * 4);
    float* dst = y + (size_t)c * HDIM + lane * 4;
    atomicAdd(dst + 0, v.x * nrm);
    atomicAdd(dst + 1, v.y * nrm);
    atomicAdd(dst + 2, v.z * nrm);
    atomicAdd(dst + 3, v.w * nrm);
}

// ---------------------------------------------------------------------------
// Epilogues: fold self-loop message + bias (+BN/ReLU | +JK max) in one pass
// ---------------------------------------------------------------------------
__global__ void k_post1(const float* __restrict__ h, float* __restrict__ y,
                        const float* __restrict__ dis,
                        const float* __restrict__ b1,
                        const float* __restrict__ gamma,
                        const float* __restrict__ beta,
                        const float* __restrict__ mean,
                        const float* __restrict__ var, int total) {
    int i = blockIdx.x * blockDim.x + threadIdx.x;
    if (i >= total) return;
    int ch = i & (HDIM - 1);
    int nd = i >> 7;
    float d = dis[nd];
    float v = y[i] + d * d * h[i] + b1[ch];              // conv out (self-loop folded)
    v = (v - mean[ch]) * rsqrtf(var[ch] + 1e-5f) * gamma[ch] + beta[ch];
    y[i] = fmaxf(v, 0.0f);                               // ReLU -> x1
}

__global__ void k_post2(const float* __restrict__ h, const float* __restrict__ x1,
                        float* __restrict__ y, const float* __restrict__ dis,
                        const float* __restrict__ b2, int total) {
    int i = blockIdx.x * blockDim.x + threadIdx.x;
    if (i >= total) return;
    int ch = i & (HDIM - 1);
    int nd = i >> 7;
    float d = dis[nd];
    float v = y[i] + d * d * h[i] + b2[ch];              // conv2 out
    y[i] = fmaxf(x1[i], v);                              // JumpingKnowledge max
}

// ---------------------------------------------------------------------------
extern "C" void kernel_launch(void* const* d_in, const int* in_sizes, int n_in,
                              void* d_out, int out_size, void* d_ws, size_t ws_size,
                              hipStream_t stream) {
    const float* node_feat = (const float*)d_in[0];
    const int*   eidx      = (const int*)d_in[1];
    const float* W1        = (const float*)d_in[2];
    const float* b1        = (const float*)d_in[3];
    const float* gamma1    = (const float*)d_in[4];
    const float* beta1     = (const float*)d_in[5];
    const float* mean1     = (const float*)d_in[6];
    const float* var1      = (const float*)d_in[7];
    const float* W2        = (const float*)d_in[8];
    const float* b2        = (const float*)d_in[9];
    const float* Wf        = (const float*)d_in[10];
    const float* bf        = (const float*)d_in[11];
    float* out = (float*)d_out;

    const int n = in_sizes[0] / HDIM;   // 100000
    const int e = in_sizes[1] / 2;      // 1600000
    const int* row = eidx;              // message sources
    const int* col = eidx + e;          // message targets

    // workspace layout (floats): dis[n] | h[n*H] | y1[n*H] | y2[n*H]
    float* dis = (float*)d_ws;
    float* h   = dis + n;
    float* y1  = h + (size_t)n * HDIM;
    float* y2  = y1 + (size_t)n * HDIM;

    const int nh      = n * HDIM;
    const int mtiles  = n >> 4;
    const int gemmGrd = (mtiles + 7) / 8;
    const int edgeGrd = (e + 7) / 8;

    // 1) symmetric norm: deg (with self loop) -> rsqrt
    k_init_deg<<<(n + 255) / 256, 256, 0, stream>>>(dis, n);
    k_edge_deg<<<(e + 255) / 256, 256, 0, stream>>>(col, dis, e);
    k_finish_deg<<<(n + 255) / 256, 256, 0, stream>>>(dis, n);

    // zero both scatter accumulators (contiguous)
    long n4 = (long)2 * nh / 4;
    k_zero4<<<(int)((n4 + 255) / 256), 256, 0, stream>>>((float4*)y1, n4);

    // 2) layer 1: h = X @ W1 (fp32 WMMA), scatter, BN+ReLU epilogue
    k_gemm_wmma<128><<<gemmGrd, 256, 0, stream>>>(node_feat, W1, nullptr, h, n, HDIM);
    k_scatter<<<edgeGrd, 256, 0, stream>>>(h, dis, row, col, y1, e);
    k_post1<<<(nh + 255) / 256, 256, 0, stream>>>(h, y1, dis, b1, gamma1, beta1, mean1, var1, nh);

    // 3) layer 2: h = x1 @ W2, scatter, bias + JK-max epilogue
    k_gemm_wmma<128><<<gemmGrd, 256, 0, stream>>>(y1, W2, nullptr, h, n, HDIM);
    k_scatter<<<edgeGrd, 256, 0, stream>>>(h, dis, row, col, y2, e);
    k_post2<<<(nh + 255) / 256, 256, 0, stream>>>(h, y1, y2, dis, b2, nh);

    // 4) final projection: out = xmax @ Wf + bf  (40 cols padded to 48)
    k_gemm_wmma<48><<<gemmGrd, 256, 0, stream>>>(y2, Wf, bf, out, n, ODIM);
}